// ContrastiveLoss_48911087567313
// MI455X (gfx1250) — compile-verified
//
#include <hip/hip_runtime.h>
#include <math.h>

typedef float v2f __attribute__((ext_vector_type(2)));
typedef float v8f __attribute__((ext_vector_type(8)));

#define NROWS 4096
#define DIM   256
#define TILE  16
#define LDS_STRIDE (DIM + 4)   // pad: 260 % 64 = 4 -> conflict-free row reads
#define INV_TEMP 2.0f          // 1 / TEMP, TEMP = 0.5
#define EPS 1e-12f

// ---------------------------------------------------------------------------
// Zero the two global accumulators (loss sum, nnz count).
__global__ void init_kernel(float* accum) {
    if (threadIdx.x < 2) accum[threadIdx.x] = 0.0f;
}

// ---------------------------------------------------------------------------
// One block per row: norm = x / max(||x||, eps)
__global__ void __launch_bounds__(DIM) normalize_kernel(const float* __restrict__ x,
                                                        float* __restrict__ y) {
    __shared__ float red[DIM];
    const int row = blockIdx.x;
    const int t = threadIdx.x;
    float v = x[(size_t)row * DIM + t];
    red[t] = v * v;
    __syncthreads();
    for (int s = DIM / 2; s > 0; s >>= 1) {
        if (t < s) red[t] += red[t + s];
        __syncthreads();
    }
    float scale = 1.0f / fmaxf(sqrtf(red[0]), EPS);
    y[(size_t)row * DIM + t] = v * scale;
}

// ---------------------------------------------------------------------------
// 16x16 sim tile via V_WMMA_F32_16X16X4_F32, K = 256.
// A strip (rows m0..m0+15) comes from LDS; B rows (n0..n0+15) from global
// (L2-resident).  Lane layout per CDNA5 ISA 7.12.2:
//   A: lane L -> M = L%16, VGPR v -> K = (L/16)*2 + v
//   B: lane L -> N = L%16, VGPR v -> K = (L/16)*2 + v
//   C/D: lane L, VGPR v -> M = v + 8*(L/16), N = L%16
__device__ inline v8f sim_tile(const float* lds_a, const float* __restrict__ norm,
                               int n0, int l16, int half) {
    v8f acc = {};
    const float* arow = lds_a + l16 * LDS_STRIDE;
    const float* brow = norm + (size_t)(n0 + l16) * DIM;
    const int kh = half * 2;
#pragma unroll 8
    for (int k = 0; k < DIM; k += 4) {
        v2f a, b;
        a.x = arow[k + kh];
        a.y = arow[k + kh + 1];
        b.x = brow[k + kh];
        b.y = brow[k + kh + 1];
        acc = __builtin_amdgcn_wmma_f32_16x16x4_f32(
            /*neg_a=*/false, a, /*neg_b=*/false, b,
            /*c_mod=*/(short)0, acc, /*reuse_a=*/false, /*reuse_b=*/false);
    }
    return acc;
}

// ---------------------------------------------------------------------------
// Pass 1: per-row sum of exp(2*sim) over different-label columns.
// One workgroup (8 waves) per 16-row M strip; waves split the 256 N tiles.
__global__ void __launch_bounds__(256) pass1_kernel(const float* __restrict__ norm,
                                                    const int* __restrict__ labels,
                                                    float* __restrict__ neg_sum) {
    __shared__ float lds_a[TILE * LDS_STRIDE];
    __shared__ float part[8][TILE];

    const int m0 = blockIdx.x * TILE;
    const int tid = threadIdx.x;

    for (int idx = tid; idx < TILE * DIM; idx += 256) {
        int r = idx >> 8, c = idx & (DIM - 1);
        lds_a[r * LDS_STRIDE + c] = norm[(size_t)(m0 + r) * DIM + c];
    }
    __syncthreads();

    const int wave = tid >> 5;
    const int lane = tid & 31;
    const int half = lane >> 4;
    const int l16  = lane & 15;

    int lblM[8];
#pragma unroll
    for (int v = 0; v < 8; ++v) lblM[v] = labels[m0 + v + half * 8];

    float negacc[8];
#pragma unroll
    for (int v = 0; v < 8; ++v) negacc[v] = 0.0f;

    for (int nj = wave; nj < NROWS / TILE; nj += 8) {
        const int n0 = nj * TILE;
        v8f acc = sim_tile(lds_a, norm, n0, l16, half);
        const int n = n0 + l16;
        const int lblN = labels[n];
#pragma unroll
        for (int v = 0; v < 8; ++v) {
            const int m = m0 + v + half * 8;
            float e = (m == n) ? 0.0f : __expf(INV_TEMP * acc[v]);
            if (lblM[v] != lblN) negacc[v] += e;
        }
    }

    // reduce over the 16 lanes (N positions) sharing each row
#pragma unroll
    for (int v = 0; v < 8; ++v) {
        float s = negacc[v];
        for (int off = 8; off > 0; off >>= 1) s += __shfl_xor(s, off, 32);
        if (l16 == 0) part[wave][v + half * 8] = s;
    }
    __syncthreads();

    if (tid < TILE) {
        float s = 0.0f;
#pragma unroll
        for (int w = 0; w < 8; ++w) s += part[w][tid];
        neg_sum[m0 + tid] = s;
    }
}

// ---------------------------------------------------------------------------
// Pass 2: loss terms for same-label off-diagonal pairs:
//   term = -log(e/(e+neg)) = log(e + neg_sum[m]) - 2*sim
__global__ void __launch_bounds__(256) pass2_kernel(const float* __restrict__ norm,
                                                    const int* __restrict__ labels,
                                                    const float* __restrict__ neg_sum,
                                                    float* __restrict__ accum) {
    __shared__ float lds_a[TILE * LDS_STRIDE];

    const int m0 = blockIdx.x * TILE;
    const int tid = threadIdx.x;

    for (int idx = tid; idx < TILE * DIM; idx += 256) {
        int r = idx >> 8, c = idx & (DIM - 1);
        lds_a[r * LDS_STRIDE + c] = norm[(size_t)(m0 + r) * DIM + c];
    }
    __syncthreads();

    const int wave = tid >> 5;
    const int lane = tid & 31;
    const int half = lane >> 4;
    const int l16  = lane & 15;

    int   lblM[8];
    float negM[8];
#pragma unroll
    for (int v = 0; v < 8; ++v) {
        lblM[v] = labels[m0 + v + half * 8];
        negM[v] = neg_sum[m0 + v + half * 8];
    }

    float lossacc = 0.0f;
    float cntacc  = 0.0f;

    for (int nj = wave; nj < NROWS / TILE; nj += 8) {
        const int n0 = nj * TILE;
        v8f acc = sim_tile(lds_a, norm, n0, l16, half);
        const int n = n0 + l16;
        const int lblN = labels[n];
#pragma unroll
        for (int v = 0; v < 8; ++v) {
            const int m = m0 + v + half * 8;
            if (lblM[v] == lblN && m != n) {
                float s2 = INV_TEMP * acc[v];
                float e = __expf(s2);
                float term = logf(e + negM[v]) - s2;
                lossacc += term;
                cntacc += (term != 0.0f) ? 1.0f : 0.0f;
            }
        }
    }

    // full-wave reduction, then one atomic per wave
    for (int off = 16; off > 0; off >>= 1) {
        lossacc += __shfl_xor(lossacc, off, 32);
        cntacc  += __shfl_xor(cntacc,  off, 32);
    }
    if (lane == 0) {
        atomicAdd(&accum[0], lossacc);
        atomicAdd(&accum[1], cntacc);
    }
}

// ---------------------------------------------------------------------------
__global__ void finalize_kernel(const float* __restrict__ accum, float* __restrict__ out) {
    if (threadIdx.x == 0) out[0] = accum[0] / accum[1];
}

// ---------------------------------------------------------------------------
extern "C" void kernel_launch(void* const* d_in, const int* in_sizes, int n_in,
                              void* d_out, int out_size, void* d_ws, size_t ws_size,
                              hipStream_t stream) {
    const float* x      = (const float*)d_in[0];
    const int*   labels = (const int*)d_in[1];
    float* out = (float*)d_out;

    // workspace layout: normalized matrix | neg_sum | accumulators
    float* norm    = (float*)d_ws;             // 4096*256 floats (4 MB)
    float* neg_sum = norm + (size_t)NROWS * DIM; // 4096 floats
    float* accum   = neg_sum + NROWS;          // 2 floats: loss sum, nnz

    init_kernel<<<1, 32, 0, stream>>>(accum);
    normalize_kernel<<<NROWS, DIM, 0, stream>>>(x, norm);
    pass1_kernel<<<NROWS / TILE, 256, 0, stream>>>(norm, labels, neg_sum);
    pass2_kernel<<<NROWS / TILE, 256, 0, stream>>>(norm, labels, neg_sum, accum);
    finalize_kernel<<<1, 32, 0, stream>>>(accum, out);
}